// Quantizer_4922032521815
// MI455X (gfx1250) — compile-verified
//
#include <hip/hip_runtime.h>

// Problem constants (match reference)
#define V_CODES 32000
#define EMB     4096
#define DIM     256
#define NROWS   8192
#define WORDS   8
#define CHUNK   1024   // rows of z_e per dist chunk (131 MB scratch)

#define KBLK    32     // K elements staged per LDS block
#define LDS_PAD 36     // floats per LDS tile row (16B-aligned rows, bank-conflict-free)

typedef __attribute__((ext_vector_type(2))) float v2f;
typedef __attribute__((ext_vector_type(4))) float v4f;
typedef __attribute__((ext_vector_type(8))) float v8f;

// D = A(16x4) * B(4x16) + C, exact fp32 matrix pipe on gfx1250.
__device__ __forceinline__ v8f wmma_f32(v2f a, v2f b, v8f c) {
  return __builtin_amdgcn_wmma_f32_16x16x4_f32(false, a, false, b, (short)0, c,
                                               false, false);
}

// CDNA5 async DMA: 16B per lane, global -> LDS, tracked by ASYNCcnt.
__device__ __forceinline__ void async_copy_b128(unsigned lds_off, const float* g) {
  asm volatile("global_load_async_to_lds_b128 %0, %1, off"
               :: "v"(lds_off), "v"(g) : "memory");
}

__device__ __forceinline__ void wait_async0() {
  asm volatile("s_wait_asynccnt 0" ::: "memory");
}

// Issue one 128x32 f32 tile pair (A,B) into LDS. 256 threads: thread t copies
// rows (t>>3)+{0,32,64,96}, 16B column chunk (t&7)*4. 8 async issues per wave.
__device__ __forceinline__ void issue_tile_copy(
    const float* gA, const float* gB, unsigned ldsA, unsigned ldsB,
    int K, int koff) {
#pragma unroll
  for (int i = 0; i < 4; ++i) {
    async_copy_b128(ldsA + (unsigned)(i * 32 * LDS_PAD * 4),
                    gA + (size_t)i * 32 * K + koff);
    async_copy_b128(ldsB + (unsigned)(i * 32 * LDS_PAD * 4),
                    gB + (size_t)i * 32 * K + koff);
  }
}

// C[m,n] = sum_k A[m,k]*B[n,k] (+ bias[n]); A:[MxK] B:[NxK] C:[MxN], row-major.
// Workgroup tile 128x128 (8 waves, 4 row-slices x 2 col-slices, wave tile 32x64).
// K staged through double-buffered LDS in blocks of 32 via async-to-LDS DMA.
// M,N multiples of 128; K multiple of 32.
template <bool BIAS>
__global__ __launch_bounds__(256) void wmma_gemm_nt(
    const float* __restrict__ A, const float* __restrict__ B,
    const float* __restrict__ bias, float* __restrict__ C,
    int M, int N, int K) {
  __shared__ float sA[2][128 * LDS_PAD];
  __shared__ float sB[2][128 * LDS_PAD];

  const int tid  = threadIdx.x;
  const int tilesM = M >> 7;
  const int tm   = blockIdx.x % tilesM;
  const int tn   = blockIdx.x / tilesM;
  const int wave = tid >> 5;
  const int lane = tid & 31;
  const int wr   = wave & 3;              // 32-row slice within tile
  const int wc   = wave >> 2;             // 64-col slice within tile
  const int r    = lane & 15;             // fragment row/col within 16
  const int kk   = (lane >> 4) << 1;      // K sub-offset: 0 or 2

  const int m00 = tm << 7;
  const int n00 = tn << 7;

  // Copy-thread mapping
  const int crow = tid >> 3;              // 0..31
  const int ccol = (tid & 7) << 2;        // float offset 0,4,..,28
  const float* gA = A + (size_t)(m00 + crow) * K + ccol;
  const float* gB = B + (size_t)(n00 + crow) * K + ccol;
  const unsigned cidx = (unsigned)(crow * LDS_PAD + ccol) * 4u;
  const unsigned a_ls0 = (unsigned)(uintptr_t)(&sA[0][0]) + cidx;
  const unsigned a_ls1 = (unsigned)(uintptr_t)(&sA[1][0]) + cidx;
  const unsigned b_ls0 = (unsigned)(uintptr_t)(&sB[0][0]) + cidx;
  const unsigned b_ls1 = (unsigned)(uintptr_t)(&sB[1][0]) + cidx;

  v8f acc[2][4];
#pragma unroll
  for (int i = 0; i < 2; ++i)
#pragma unroll
    for (int j = 0; j < 4; ++j)
#pragma unroll
      for (int e = 0; e < 8; ++e) acc[i][j][e] = 0.0f;

  // Fragment base indices within an LDS tile (float units)
  const int arow = (wr << 5) + r;         // A rows: arow, arow+16
  const int brow = (wc << 6) + r;         // B cols: brow + {0,16,32,48}

  const int nblk = K >> 5;
  issue_tile_copy(gA, gB, a_ls0, b_ls0, K, 0);

  for (int kb = 0; kb < nblk; ++kb) {
    wait_async0();        // my block-kb copies have landed in LDS
    __syncthreads();      // everyone's copies visible; prev compute finished
    if (kb + 1 < nblk) {
      const bool odd = (kb + 1) & 1;
      issue_tile_copy(gA, gB, odd ? a_ls1 : a_ls0, odd ? b_ls1 : b_ls0,
                      K, (kb + 1) << 5);
    }
    const float* As = sA[kb & 1];
    const float* Bs = sB[kb & 1];
#pragma unroll
    for (int ks = 0; ks < KBLK; ks += 4) {
      v2f a0 = *(const v2f*)&As[arow * LDS_PAD + ks + kk];
      v2f a1 = *(const v2f*)&As[(arow + 16) * LDS_PAD + ks + kk];
      v2f b0 = *(const v2f*)&Bs[brow * LDS_PAD + ks + kk];
      v2f b1 = *(const v2f*)&Bs[(brow + 16) * LDS_PAD + ks + kk];
      v2f b2 = *(const v2f*)&Bs[(brow + 32) * LDS_PAD + ks + kk];
      v2f b3 = *(const v2f*)&Bs[(brow + 48) * LDS_PAD + ks + kk];
      acc[0][0] = wmma_f32(a0, b0, acc[0][0]);
      acc[0][1] = wmma_f32(a0, b1, acc[0][1]);
      acc[0][2] = wmma_f32(a0, b2, acc[0][2]);
      acc[0][3] = wmma_f32(a0, b3, acc[0][3]);
      acc[1][0] = wmma_f32(a1, b0, acc[1][0]);
      acc[1][1] = wmma_f32(a1, b1, acc[1][1]);
      acc[1][2] = wmma_f32(a1, b2, acc[1][2]);
      acc[1][3] = wmma_f32(a1, b3, acc[1][3]);
    }
    __syncthreads();      // all waves done reading this buffer before refill
  }

  const int m0 = m00 + (wr << 5);
  const int n0 = n00 + (wc << 6);
  const int rowoff = (lane >> 4) << 3;    // 0 or 8
#pragma unroll
  for (int j = 0; j < 4; ++j) {
    const int col = n0 + (j << 4) + r;
    const float bb = BIAS ? bias[col] : 0.0f;
#pragma unroll
    for (int i = 0; i < 2; ++i) {
      float* cp = C + (size_t)(m0 + (i << 4) + rowoff) * N + col;
#pragma unroll
      for (int e = 0; e < 8; ++e) cp[(size_t)e * N] = acc[i][j][e] + bb;
    }
  }
}

// msq[v] = sum_d mapped[v,d]^2 ; one wave per row, DIM==256.
__global__ __launch_bounds__(256) void rowsq_kernel(
    const float* __restrict__ mapped, float* __restrict__ msq) {
  const int wave = threadIdx.x >> 5;
  const int lane = threadIdx.x & 31;
  const int row  = blockIdx.x * 8 + wave;
  const float* p = mapped + (size_t)row * DIM + lane * 8;
  v4f x0 = *(const v4f*)p;
  v4f x1 = *(const v4f*)(p + 4);
  float s = 0.0f;
#pragma unroll
  for (int e = 0; e < 4; ++e) s += x0[e] * x0[e] + x1[e] * x1[e];
#pragma unroll
  for (int off = 16; off; off >>= 1) s += __shfl_xor(s, off, 32);
  if (lane == 0) msq[row] = s;
}

// Greedy masked argmin: one block per batch row (8 consecutive z_e rows).
// score[n] = msq[n] - 2*dot[row][n]  (row-constant ||z_e||^2 is argmin-invariant)
__global__ __launch_bounds__(256) void argmin_kernel(
    const float* __restrict__ dot, const float* __restrict__ msq,
    int* __restrict__ indices, int row_base) {
  __shared__ int   picked[WORDS];
  __shared__ float wval[8];
  __shared__ int   widx[8];
  const int tid  = threadIdx.x;
  const int lane = tid & 31;
  const int wave = tid >> 5;
  const int lb   = blockIdx.x;  // local batch row within chunk

  for (int i = 0; i < WORDS; ++i) {
    int pk[WORDS - 1];
    for (int t = 0; t < i; ++t) pk[t] = picked[t];

    const float* drow = dot + (size_t)(lb * WORDS + i) * V_CODES;
    float best  = 3.402823466e+38f;
    int   bestn = 0x7fffffff;
    for (int n = tid; n < V_CODES; n += 256) {
      float v = msq[n] - 2.0f * drow[n];
      bool masked = false;
      for (int t = 0; t < i; ++t) masked |= (pk[t] == n);
      if (!masked && (v < best || (v == best && n < bestn))) { best = v; bestn = n; }
    }
#pragma unroll
    for (int off = 16; off; off >>= 1) {
      float ov = __shfl_xor(best, off, 32);
      int   on = __shfl_xor(bestn, off, 32);
      if (ov < best || (ov == best && on < bestn)) { best = ov; bestn = on; }
    }
    if (lane == 0) { wval[wave] = best; widx[wave] = bestn; }
    __syncthreads();
    if (tid == 0) {
      float bb = wval[0]; int bn = widx[0];
      for (int w = 1; w < 8; ++w)
        if (wval[w] < bb || (wval[w] == bb && widx[w] < bn)) { bb = wval[w]; bn = widx[w]; }
      picked[i] = bn;
      indices[row_base + lb * WORDS + i] = bn;
    }
    __syncthreads();
  }
}

// z_q[n,:] = mapped[idx[n],:]; partials[n] = sum_d (z_q - z_e)^2. DIM==blockDim.
__global__ __launch_bounds__(256) void gather_loss_kernel(
    const float* __restrict__ mapped, const float* __restrict__ z_e,
    const int* __restrict__ indices, float* __restrict__ z_q,
    float* __restrict__ partials) {
  const int n = blockIdx.x;
  const int t = threadIdx.x;
  const int idx = indices[n];
  const float q = mapped[(size_t)idx * DIM + t];
  const float e = z_e[(size_t)n * DIM + t];
  z_q[(size_t)n * DIM + t] = q;
  float d = q - e;
  float s = d * d;
  const int lane = t & 31, wave = t >> 5;
#pragma unroll
  for (int off = 16; off; off >>= 1) s += __shfl_xor(s, off, 32);
  __shared__ float sm[8];
  if (lane == 0) sm[wave] = s;
  __syncthreads();
  if (t == 0) {
    float tot = 0.0f;
    for (int w = 0; w < 8; ++w) tot += sm[w];
    partials[n] = tot;
  }
}

// Deterministic final reduction: loss = mean((z_q - z_e)^2)
// (0.75*vq + 0.25*commit == mean; both terms identical in forward)
__global__ __launch_bounds__(256) void finalize_loss(
    const float* __restrict__ partials, float* __restrict__ out) {
  __shared__ float sm[256];
  float s = 0.0f;
  for (int i = threadIdx.x; i < NROWS; i += 256) s += partials[i];
  sm[threadIdx.x] = s;
  __syncthreads();
  for (int stride = 128; stride > 0; stride >>= 1) {
    if (threadIdx.x < stride) sm[threadIdx.x] += sm[threadIdx.x + stride];
    __syncthreads();
  }
  if (threadIdx.x == 0) *out = sm[0] * (1.0f / ((float)NROWS * (float)DIM));
}

extern "C" void kernel_launch(void* const* d_in, const int* in_sizes, int n_in,
                              void* d_out, int out_size, void* d_ws, size_t ws_size,
                              hipStream_t stream) {
  const float* z_e      = (const float*)d_in[0];
  const float* codebook = (const float*)d_in[1];
  const float* W        = (const float*)d_in[2];
  const float* b        = (const float*)d_in[3];
  (void)d_in[4];  // word_num == 8, fixed at compile time
  (void)in_sizes; (void)n_in; (void)out_size; (void)ws_size;

  // Workspace layout (all offsets 256B-aligned): ~165 MB total
  char* ws = (char*)d_ws;
  float* mapped   = (float*)(ws);                                       // V*D f32
  float* msq      = (float*)(ws + (size_t)V_CODES * DIM * 4);           // V f32
  float* dot      = (float*)(ws + (size_t)V_CODES * DIM * 4
                                + (size_t)V_CODES * 4);                 // CHUNK*V f32
  int*   indices  = (int*)  (ws + (size_t)V_CODES * DIM * 4
                                + (size_t)V_CODES * 4
                                + (size_t)CHUNK * V_CODES * 4);         // N i32
  float* partials = (float*)(ws + (size_t)V_CODES * DIM * 4
                                + (size_t)V_CODES * 4
                                + (size_t)CHUNK * V_CODES * 4
                                + (size_t)NROWS * 4);                   // N f32
  float* z_q  = (float*)d_out;
  float* loss = z_q + (size_t)NROWS * DIM;

  // 1) mapped = codebook @ W.T + b   (M=32000, N=256, K=4096)
  wmma_gemm_nt<true><<<dim3((V_CODES / 128) * (DIM / 128)), dim3(256), 0, stream>>>(
      codebook, W, b, mapped, V_CODES, DIM, EMB);

  // 2) msq[v] = ||mapped[v]||^2
  rowsq_kernel<<<dim3(V_CODES / 8), dim3(256), 0, stream>>>(mapped, msq);

  // 3) per 1024-row chunk: dot = z_e_chunk @ mapped.T, then greedy masked argmin
  for (int c = 0; c < NROWS / CHUNK; ++c) {
    wmma_gemm_nt<false><<<dim3((CHUNK / 128) * (V_CODES / 128)), dim3(256), 0, stream>>>(
        z_e + (size_t)c * CHUNK * DIM, mapped, nullptr, dot, CHUNK, V_CODES, DIM);
    argmin_kernel<<<dim3(CHUNK / WORDS), dim3(256), 0, stream>>>(
        dot, msq, indices, c * CHUNK);
  }

  // 4) gather z_q + per-row squared error
  gather_loss_kernel<<<dim3(NROWS), dim3(256), 0, stream>>>(
      mapped, z_e, indices, z_q, partials);

  // 5) scalar loss
  finalize_loss<<<dim3(1), dim3(256), 0, stream>>>(partials, loss);
}